// VladPooling_21629455302857
// MI455X (gfx1250) — compile-verified
//
#include <hip/hip_runtime.h>
#include <math.h>

typedef __attribute__((ext_vector_type(2))) float v2f;
typedef __attribute__((ext_vector_type(8))) float v8f;

#define B_    32
#define T_    500
#define D_    512
#define KC    10    // total centers (K_CENTERS + G_CENTERS)
#define KOUT  8     // K_CENTERS kept in output
#define KPAD  16    // pad clusters to WMMA M=16

// ---------------------------------------------------------------------------
// Kernel 1: softmax over the 10 cluster scores per (b,t); write A padded to 16
// (zeros in k=10..15 so the WMMA M-padding rows accumulate exact zeros).
// ---------------------------------------------------------------------------
__global__ void vlad_softmax(const float* __restrict__ score,
                             float* __restrict__ Apad) {
  int idx = blockIdx.x * blockDim.x + threadIdx.x;
  if (idx >= B_ * T_) return;
  const float* s = score + (size_t)idx * KC;
  float m = s[0];
#pragma unroll
  for (int k = 1; k < KC; ++k) m = fmaxf(m, s[k]);
  float e[KC];
  float sum = 0.f;
#pragma unroll
  for (int k = 0; k < KC; ++k) { e[k] = __expf(s[k] - m); sum += e[k]; }
  float inv = 1.0f / sum;
  float* o = Apad + (size_t)idx * KPAD;
#pragma unroll
  for (int k = 0; k < KC; ++k) o[k] = e[k] * inv;
#pragma unroll
  for (int k = KC; k < KPAD; ++k) o[k] = 0.f;
}

// ---------------------------------------------------------------------------
// Kernel 2: a_sum[b][k] = sum_t A[b][t][k]  (one wave per k, wave32 shuffle
// reduction). Only k < 8 are consumed downstream.
// ---------------------------------------------------------------------------
__global__ void vlad_asum(const float* __restrict__ Apad,
                          float* __restrict__ a_sum) {
  int b    = blockIdx.x;
  int wave = threadIdx.x >> 5;   // 8 waves -> k = 0..7
  int lane = threadIdx.x & 31;
  float s = 0.f;
  for (int t = lane; t < T_; t += 32)
    s += Apad[((size_t)b * T_ + t) * KPAD + wave];
#pragma unroll
  for (int off = 16; off > 0; off >>= 1) s += __shfl_xor(s, off, 32);
  if (lane == 0) a_sum[b * KPAD + wave] = s;
}

// ---------------------------------------------------------------------------
// Kernel 3: per-batch GEMM  agg[b] = A_b^T (16x500, padded) x feat_b (500x512)
// via V_WMMA_F32_16X16X4_F32. One wave owns one 16(M)x16(N) tile and walks
// K (=time) in steps of 4. grid = (B, 4), block = 256 (8 waves -> 8 N-tiles).
//
// A 16x4 f32 fragment (2 VGPRs, per ISA 7.12.2):
//   lane L: M = L&15 ; VGPR0 holds K = (L>>4)*2, VGPR1 holds K = (L>>4)*2+1
// B 4x16 f32 fragment mirrors it with lane = N.
// C/D 16x16 f32: VGPR v, lane L -> M = v + (L>>4)*8, N = L&15.
// ---------------------------------------------------------------------------
__global__ void vlad_gemm(const float* __restrict__ feat,
                          const float* __restrict__ Apad,
                          float* __restrict__ agg) {
  int b    = blockIdx.x;
  int wave = threadIdx.x >> 5;
  int lane = threadIdx.x & 31;
  int n0   = (blockIdx.y * 8 + wave) * 16;   // N-tile origin in D
  int m    = lane & 15;                      // M for A-frag, N for B-frag
  int kh   = (lane >> 4) * 2;                // K sub-offset for this lane half

  const float* Ab = Apad + (size_t)b * T_ * KPAD;
  const float* Fb = feat + (size_t)b * T_ * D_;

  v8f c = {};
  for (int t0 = 0; t0 < T_; t0 += 4) {       // 125 iterations, exact
    v2f a, bf;
    a.x  = Ab[(size_t)(t0 + kh    ) * KPAD + m];
    a.y  = Ab[(size_t)(t0 + kh + 1) * KPAD + m];
    bf.x = Fb[(size_t)(t0 + kh    ) * D_ + n0 + m];
    bf.y = Fb[(size_t)(t0 + kh + 1) * D_ + n0 + m];
    // 8 args: (neg_a, A, neg_b, B, c_mod, C, reuse_a, reuse_b)
    c = __builtin_amdgcn_wmma_f32_16x16x4_f32(false, a, false, bf,
                                              (short)0, c, false, false);
  }

#pragma unroll
  for (int v = 0; v < 8; ++v) {
    int M = v + (lane >> 4) * 8;
    agg[((size_t)b * KPAD + M) * D_ + n0 + m] = c[v];
  }
}

// ---------------------------------------------------------------------------
// Kernel 4: residual + L2 normalize.  One block per (b, k<8); 256 threads,
// 2 elements of D=512 each; LDS tree reduction for the squared norm.
// ---------------------------------------------------------------------------
__global__ void vlad_finalize(const float* __restrict__ agg,
                              const float* __restrict__ a_sum,
                              const float* __restrict__ cluster,
                              float* __restrict__ out) {
  int b   = blockIdx.x >> 3;
  int k   = blockIdx.x & 7;
  int tid = threadIdx.x;

  float as  = a_sum[b * KPAD + k];
  size_t base = ((size_t)b * KPAD + k) * D_;
  float r0 = agg[base + tid]       - as * cluster[(size_t)k * D_ + tid];
  float r1 = agg[base + tid + 256] - as * cluster[(size_t)k * D_ + tid + 256];

  __shared__ float red[256];
  red[tid] = r0 * r0 + r1 * r1;
  __syncthreads();
#pragma unroll
  for (int s = 128; s > 0; s >>= 1) {
    if (tid < s) red[tid] += red[tid + s];
    __syncthreads();
  }
  __shared__ float inv;
  if (tid == 0) inv = rsqrtf(fmaxf(red[0], 1e-12f));
  __syncthreads();

  size_t ob = ((size_t)b * KOUT + k) * D_;
  out[ob + tid]       = r0 * inv;
  out[ob + tid + 256] = r1 * inv;
}

// ---------------------------------------------------------------------------
extern "C" void kernel_launch(void* const* d_in, const int* in_sizes, int n_in,
                              void* d_out, int out_size, void* d_ws, size_t ws_size,
                              hipStream_t stream) {
  const float* feat    = (const float*)d_in[0];  // (32,1,500,512)
  const float* score   = (const float*)d_in[1];  // (32,1,500,10)
  const float* cluster = (const float*)d_in[2];  // (10,512)
  float* out = (float*)d_out;                    // (32, 8*512)

  float* ws    = (float*)d_ws;
  float* Apad  = ws;                             // 32*500*16 = 256000 floats
  float* a_sum = Apad + (size_t)B_ * T_ * KPAD;  // 32*16     = 512 floats
  float* agg   = a_sum + (size_t)B_ * KPAD;      // 32*16*512 = 262144 floats

  vlad_softmax<<<(B_ * T_ + 255) / 256, 256, 0, stream>>>(score, Apad);
  vlad_asum<<<B_, 256, 0, stream>>>(Apad, a_sum);
  vlad_gemm<<<dim3(B_, 4), 256, 0, stream>>>(feat, Apad, agg);
  vlad_finalize<<<B_ * KOUT, 256, 0, stream>>>(agg, a_sum, cluster, out);
}